// GraphConvolution_77524159693609
// MI455X (gfx1250) — compile-verified
//
#include <hip/hip_runtime.h>
#include <hip/hip_bf16.h>
#include <math.h>

// ---------------------------------------------------------------------------
// GraphConvolution (hyperbolic GCN layer) for MI455X / gfx1250, wave32.
//   inputs : adjacency [N,N] f32, input_feature [N,128] f32,
//            weight [128,64] f32, bias [64] f32
//   output : relu(adjacency @ logmap0(proj(mobius_add(proj-free support, hyp_bias)))) [N,64] f32
// Kernel 1: per-row hyperbolic math (VALU, one wave per row) -> d_ws
// Kernel 2: 16384x16384x64 f32 GEMM with V_WMMA_F32_16X16X4_F32 + fused ReLU
// ---------------------------------------------------------------------------

typedef __attribute__((ext_vector_type(2))) float v2f;
typedef __attribute__((ext_vector_type(8))) float v8f;

#define GC_N     16384
#define GC_DIN   128
#define GC_DOUT  64
#define MIN_NORM 1e-15f
#define MAXNORM  (1.0f - 4e-3f)   // (1 - BALL_EPS)/sqrt(c), c = 1

static __device__ __forceinline__ float wave_sum(float v) {
  // full wave32 reduction
  #pragma unroll
  for (int off = 16; off > 0; off >>= 1) v += __shfl_xor(v, off, 32);
  return v;
}

static __device__ __forceinline__ float artanh_c(float x) {
  // torch Artanh clamps away from +-1
  x = fminf(fmaxf(x, -1.0f + 1e-7f), 1.0f - 1e-7f);
  return 0.5f * __logf((1.0f + x) / (1.0f - x));
}

// ---------------------------------------------------------------------------
// Kernel 1: support_tangent[row] = logmap0(proj(mobius_add(mobius_matvec(W,x), hyp_bias)))
// blockDim = 256 (8 waves), one row per wave.
// ---------------------------------------------------------------------------
__global__ __launch_bounds__(256) void hgcn_rowmath_kernel(
    const float* __restrict__ X, const float* __restrict__ W,
    const float* __restrict__ bias, float* __restrict__ St) {
  __shared__ float xs[8][GC_DIN];
  const int lane = threadIdx.x & 31;
  const int wv   = threadIdx.x >> 5;
  const int row  = blockIdx.x * 8 + wv;

  // ---- load x row (4 floats / lane), accumulate ||x||^2 -------------------
  float4 x4 = *(const float4*)(X + (size_t)row * GC_DIN + lane * 4);
  *(float4*)&xs[wv][lane * 4] = x4;
  float xn2 = wave_sum(x4.x * x4.x + x4.y * x4.y + x4.z * x4.z + x4.w * x4.w);
  __syncthreads();   // uniform across block (grid covers N exactly)

  // ---- mx = x @ W : lane owns output cols {lane, lane+32} -----------------
  float mx0 = 0.0f, mx1 = 0.0f;
  #pragma unroll 4
  for (int k = 0; k < GC_DIN; ++k) {
    float xv = xs[wv][k];
    mx0 = fmaf(xv, W[k * GC_DOUT + lane], mx0);
    mx1 = fmaf(xv, W[k * GC_DOUT + lane + 32], mx1);
  }

  // ---- mobius_matvec tail -------------------------------------------------
  float x_norm  = fmaxf(sqrtf(xn2), MIN_NORM);
  float mx_nrm  = fmaxf(sqrtf(wave_sum(mx0 * mx0 + mx1 * mx1)), MIN_NORM);
  float t       = tanhf(mx_nrm / x_norm * artanh_c(x_norm)) / mx_nrm;
  float s0 = t * mx0, s1 = t * mx1;
  unsigned long long nz = __ballot(mx0 != 0.0f || mx1 != 0.0f);
  if (nz == 0ull) { s0 = 0.0f; s1 = 0.0f; }

  // ---- hyp_bias = proj(expmap0(bias)) (redundant per wave, tiny) ----------
  float b0 = bias[lane], b1 = bias[lane + 32];
  float un = fmaxf(sqrtf(wave_sum(b0 * b0 + b1 * b1)), MIN_NORM);
  float tb = tanhf(un) / un;
  float h0 = tb * b0, h1 = tb * b1;
  float hn = fmaxf(sqrtf(wave_sum(h0 * h0 + h1 * h1)), MIN_NORM);
  if (hn > MAXNORM) { float sc = MAXNORM / hn; h0 *= sc; h1 *= sc; }

  // ---- mobius_add(support, hyp_bias) --------------------------------------
  float x2 = wave_sum(s0 * s0 + s1 * s1);
  float y2 = wave_sum(h0 * h0 + h1 * h1);
  float xy = wave_sum(s0 * h0 + s1 * h1);
  float cx = 1.0f + 2.0f * xy + y2;
  float cy = 1.0f - x2;
  float dn = fmaxf(1.0f + 2.0f * xy + x2 * y2, MIN_NORM);
  float p0 = (cx * s0 + cy * h0) / dn;
  float p1 = (cx * s1 + cy * h1) / dn;

  // ---- proj ---------------------------------------------------------------
  float pn = fmaxf(sqrtf(wave_sum(p0 * p0 + p1 * p1)), MIN_NORM);
  if (pn > MAXNORM) { float sc = MAXNORM / pn; p0 *= sc; p1 *= sc; pn = MAXNORM; }

  // ---- logmap0 ------------------------------------------------------------
  float lt = artanh_c(pn) / pn;
  St[(size_t)row * GC_DOUT + lane]      = lt * p0;
  St[(size_t)row * GC_DOUT + lane + 32] = lt * p1;
}

// ---------------------------------------------------------------------------
// Kernel 2: out = relu(A @ B), A = adjacency [N,N], B = support_tangent [N,64]
// grid = N/16 workgroups, block = 128 (4 waves). Wave w -> cols [16w,16w+16).
// A panel (16 rows x 64 K, fp32, 4 KB) staged in LDS with coalesced loads.
// WMMA f32 16x16x4: A frag lane layout (ISA 7.12.2): lanes 0-15 K={kb,kb+1},
// lanes 16-31 K={kb+2,kb+3}, M = lane%16. B mirrored: N = lane%16.
// C/D layout: VGPR r holds M = r (lanes 0-15) / r+8 (lanes 16-31), N = lane%16.
// ---------------------------------------------------------------------------
#define TK 64
#define APAD 68   // LDS row stride in floats: 16B aligned, conflict-free reads

__global__ __launch_bounds__(128) void hgcn_aggregate_wmma_kernel(
    const float* __restrict__ A, const float* __restrict__ B,
    float* __restrict__ out) {
  __shared__ float Alds[16 * APAD];
  const int tid   = threadIdx.x;
  const int wave  = tid >> 5;
  const int lane  = tid & 31;
  const int m     = lane & 15;        // M index (A) / N index (B, C)
  const int khalf = lane >> 4;        // 0: K pair {0,1}, 1: K pair {2,3}
  const int row0  = blockIdx.x * 16;

  // cooperative A-panel load: 8 threads per row, 2 x float4 each (256 B/row)
  const int tr = tid >> 3, tc = tid & 7;
  const float* Asrc = A + (size_t)(row0 + tr) * GC_N + (size_t)tc * 8;
  float* Adst = &Alds[tr * APAD + tc * 8];

  v8f c = {0.f, 0.f, 0.f, 0.f, 0.f, 0.f, 0.f, 0.f};

  for (int kb = 0; kb < GC_N; kb += TK) {
    float4 a0 = ((const float4*)(Asrc + kb))[0];
    float4 a1 = ((const float4*)(Asrc + kb))[1];
    __syncthreads();                      // previous panel fully consumed
    ((float4*)Adst)[0] = a0;
    ((float4*)Adst)[1] = a1;
    __syncthreads();                      // panel visible to all waves

    const float* Bp = B + (size_t)kb * GC_DOUT + wave * 16 + m;
    #pragma unroll
    for (int kk = 0; kk < TK; kk += 4) {
      const int ka = kk + khalf * 2;
      v2f a = *(const v2f*)&Alds[m * APAD + ka];          // A[m][ka], A[m][ka+1]
      v2f b;
      b.x = Bp[(size_t)ka * GC_DOUT];                     // B[kb+ka  ][n]
      b.y = Bp[(size_t)(ka + 1) * GC_DOUT];               // B[kb+ka+1][n]
      c = __builtin_amdgcn_wmma_f32_16x16x4_f32(
              /*neg_a=*/false, a, /*neg_b=*/false, b,
              /*c_mod=*/(short)0, c, /*reuse_a=*/false, /*reuse_b=*/false);
    }
  }

  // fused ReLU + store (coalesced: 16 lanes x 4 B contiguous per row)
  float* Op = out + (size_t)row0 * GC_DOUT + wave * 16 + m;
  #pragma unroll
  for (int r = 0; r < 8; ++r) {
    int rr = r + 8 * khalf;
    float v = c[r];
    Op[(size_t)rr * GC_DOUT] = v > 0.0f ? v : 0.0f;
  }
}

// ---------------------------------------------------------------------------
extern "C" void kernel_launch(void* const* d_in, const int* in_sizes, int n_in,
                              void* d_out, int out_size, void* d_ws, size_t ws_size,
                              hipStream_t stream) {
  const float* adj  = (const float*)d_in[0];   // [N, N]
  const float* X    = (const float*)d_in[1];   // [N, 128]
  const float* W    = (const float*)d_in[2];   // [128, 64]
  const float* bias = (const float*)d_in[3];   // [64]
  float* out = (float*)d_out;                  // [N, 64]
  float* St  = (float*)d_ws;                   // support_tangent [N, 64] (4 MB)

  hgcn_rowmath_kernel<<<GC_N / 8, 256, 0, stream>>>(X, W, bias, St);
  hgcn_aggregate_wmma_kernel<<<GC_N / 16, 128, 0, stream>>>(adj, St, out);
}